// GraphAttentionLayer_21474836480369
// MI455X (gfx1250) — compile-verified
//
#include <hip/hip_runtime.h>
#include <hip/hip_bf16.h>

typedef __attribute__((ext_vector_type(2))) float v2f;
typedef __attribute__((ext_vector_type(4))) float v4f;
typedef __attribute__((ext_vector_type(8))) float v8f;

#define GAT_D_IN   128
#define GAT_D_OUT  64
#define GAT_ALPHA  0.2f
// 1/sqrt(D_OUT) = 1/8
#define GAT_INVSQRT_DOUT 0.125f

__device__ __forceinline__ void atomAddF(float* p, float v) {
  __hip_atomic_fetch_add(p, v, __ATOMIC_RELAXED, __HIP_MEMORY_SCOPE_AGENT);
}

// ---------------------------------------------------------------------------
// Kernel 0: zero the accumulator (d_out used as agg) and rowsum.
// ---------------------------------------------------------------------------
__global__ void gat_zero(float* __restrict__ agg, float* __restrict__ rowsum,
                         long n_agg, long n_rs) {
  long i = (long)blockIdx.x * blockDim.x + threadIdx.x;
  if (i < n_agg) agg[i] = 0.0f;
  if (i < n_rs)  rowsum[i] = 0.0f;
}

// ---------------------------------------------------------------------------
// Kernel 1: data[N,64] = h[N,128] @ W_w[64,128]^T + W_b, via f32 WMMA 16x16x4.
// One wave per 16-row block; 4 accumulator tiles cover all 64 output cols.
// A-frag: lane L holds h[m0 + L%16][k0 + 2*(L/16) + {0,1}]   (ISA 16x4 layout)
// B-frag: lane L holds B[k0 + 2*(L/16) + {0,1}][L%16], B[k][n] = W_w[n][k]
// C/D:    lane L, VGPR r -> data[m0 + r + 8*(L/16)][16*t + L%16]
// ---------------------------------------------------------------------------
__global__ __launch_bounds__(256) void gat_gemm_wmma(
    const float* __restrict__ h, const float* __restrict__ Ww,
    const float* __restrict__ Wb, float* __restrict__ data, int N) {
  const int wave = (int)((blockIdx.x * (long)blockDim.x + threadIdx.x) >> 5);
  const int lane = (int)(threadIdx.x & 31);
  const int nblk = (N + 15) >> 4;
  if (wave >= nblk) return;                 // wave-uniform: EXEC all-1s inside
  const int m0   = wave << 4;
  const int half = lane >> 4;               // 0 or 1
  const int l16  = lane & 15;

  int mA = m0 + l16; if (mA >= N) mA = N - 1;   // clamped (tail-safe) A row
  const float* __restrict__ arow = h  + (size_t)mA * GAT_D_IN;
  const float* __restrict__ b0r  = Ww + (size_t)(l16 +  0) * GAT_D_IN;
  const float* __restrict__ b1r  = Ww + (size_t)(l16 + 16) * GAT_D_IN;
  const float* __restrict__ b2r  = Ww + (size_t)(l16 + 32) * GAT_D_IN;
  const float* __restrict__ b3r  = Ww + (size_t)(l16 + 48) * GAT_D_IN;

  v8f acc0 = {}, acc1 = {}, acc2 = {}, acc3 = {};
  #pragma unroll
  for (int k0 = 0; k0 < GAT_D_IN; k0 += 4) {
    const int kk = k0 + (half << 1);
    v2f a  = *(const v2f*)(arow + kk);
    v2f b0 = *(const v2f*)(b0r + kk);
    v2f b1 = *(const v2f*)(b1r + kk);
    v2f b2 = *(const v2f*)(b2r + kk);
    v2f b3 = *(const v2f*)(b3r + kk);
    acc0 = __builtin_amdgcn_wmma_f32_16x16x4_f32(false, a, false, b0, (short)0, acc0, false, false);
    acc1 = __builtin_amdgcn_wmma_f32_16x16x4_f32(false, a, false, b1, (short)0, acc1, false, false);
    acc2 = __builtin_amdgcn_wmma_f32_16x16x4_f32(false, a, false, b2, (short)0, acc2, false, false);
    acc3 = __builtin_amdgcn_wmma_f32_16x16x4_f32(false, a, false, b3, (short)0, acc3, false, false);
  }

  const float bb0 = Wb[l16], bb1 = Wb[l16 + 16], bb2 = Wb[l16 + 32], bb3 = Wb[l16 + 48];
  #pragma unroll
  for (int r = 0; r < 8; ++r) {
    const int m = m0 + r + (half << 3);
    if (m < N) {
      float* drow = data + (size_t)m * GAT_D_OUT;
      drow[l16 +  0] = acc0[r] + bb0;
      drow[l16 + 16] = acc1[r] + bb1;
      drow[l16 + 32] = acc2[r] + bb2;
      drow[l16 + 48] = acc3[r] + bb3;
    }
  }
}

// ---------------------------------------------------------------------------
// Kernel 2: per-node attention logits s[i] = data[i,:]·a_src, t[i] = data[i,:]·a_dst.
// One wave per node; each lane takes a float2 slice, wave-wide xor reduction.
// ---------------------------------------------------------------------------
__global__ __launch_bounds__(256) void gat_st(
    const float* __restrict__ data, const float* __restrict__ a,
    float* __restrict__ s, float* __restrict__ t, int N) {
  const int wave = (int)((blockIdx.x * (long)blockDim.x + threadIdx.x) >> 5);
  const int lane = (int)(threadIdx.x & 31);
  if (wave >= N) return;
  const float* drow = data + (size_t)wave * GAT_D_OUT;
  v2f d  = *(const v2f*)(drow + 2 * lane);
  v2f as = *(const v2f*)(a + 2 * lane);               // a_src = a[0:64]
  v2f at = *(const v2f*)(a + GAT_D_OUT + 2 * lane);   // a_dst = a[64:128]
  float ps = d.x * as.x + d.y * as.y;
  float pt = d.x * at.x + d.y * at.y;
  #pragma unroll
  for (int off = 16; off > 0; off >>= 1) {
    ps += __shfl_xor(ps, off, 32);
    pt += __shfl_xor(pt, off, 32);
  }
  if (lane == 0) { s[wave] = ps; t[wave] = pt; }
}

// ---------------------------------------------------------------------------
// Kernel 3: edge phase. One wave per edge: compute softmax weight, atomically
// accumulate rowsum[e0] and w * data[e1,:] into agg[e0,:] (L2-resident fadd).
// ---------------------------------------------------------------------------
__global__ __launch_bounds__(256) void gat_edge(
    const int* __restrict__ e0s, const int* __restrict__ e1s,
    const float* __restrict__ s, const float* __restrict__ t,
    const float* __restrict__ data, float* __restrict__ rowsum,
    float* __restrict__ agg, long E) {
  const long wave = ((long)blockIdx.x * blockDim.x + threadIdx.x) >> 5;
  const int lane  = (int)(threadIdx.x & 31);
  if (wave >= E) return;
  const int e0 = e0s[wave];
  const int e1 = e1s[wave];
  const float raw = s[e0] + t[e1];
  const float lr  = raw > 0.0f ? raw : GAT_ALPHA * raw;
  const float w   = expf(lr * GAT_INVSQRT_DOUT);
  if (lane == 0) atomAddF(&rowsum[e0], w);
  v2f d = *(const v2f*)(data + (size_t)e1 * GAT_D_OUT + 2 * lane);
  float* arow = agg + (size_t)e0 * GAT_D_OUT + 2 * lane;
  atomAddF(arow + 0, w * d.x);
  atomAddF(arow + 1, w * d.y);
}

// ---------------------------------------------------------------------------
// Kernel 4: finalize. out = (rowsum==0) ? data : agg / rowsum  (in-place on agg).
// 4 columns per thread (float4).
// ---------------------------------------------------------------------------
__global__ void gat_fin(float* __restrict__ out, const float* __restrict__ data,
                        const float* __restrict__ rowsum, int N) {
  const long i = (long)blockIdx.x * blockDim.x + threadIdx.x;
  const long node = i >> 4;
  const int  c4   = (int)(i & 15) * 4;
  if (node >= N) return;
  const float rs = rowsum[node];
  float* o = out + (size_t)node * GAT_D_OUT + c4;
  if (rs == 0.0f) {
    *(v4f*)o = *(const v4f*)(data + (size_t)node * GAT_D_OUT + c4);
  } else {
    const float inv = 1.0f / rs;
    v4f v = *(v4f*)o;
    v.x *= inv; v.y *= inv; v.z *= inv; v.w *= inv;
    *(v4f*)o = v;
  }
}

extern "C" void kernel_launch(void* const* d_in, const int* in_sizes, int n_in,
                              void* d_out, int out_size, void* d_ws, size_t ws_size,
                              hipStream_t stream) {
  const float* h  = (const float*)d_in[0];   // [N, 128]
  const int*   eg = (const int*)d_in[1];     // [2, E]
  const float* Ww = (const float*)d_in[2];   // [64, 128]
  const float* Wb = (const float*)d_in[3];   // [64]
  const float* a  = (const float*)d_in[4];   // [1, 128]

  const int  N = in_sizes[0] / GAT_D_IN;
  const long E = (long)in_sizes[1] / 2;

  float* out    = (float*)d_out;             // [N, 64], doubles as agg accumulator
  float* data   = (float*)d_ws;              // [N, 64]
  float* sv     = data + (size_t)N * GAT_D_OUT;  // [N]
  float* tv     = sv + N;                        // [N]
  float* rowsum = tv + N;                        // [N]

  {
    long n = (long)N * GAT_D_OUT;
    int blocks = (int)((n + 255) / 256);
    gat_zero<<<blocks, 256, 0, stream>>>(out, rowsum, n, (long)N);
  }
  {
    int nblk = (N + 15) >> 4;                // 16-row blocks (waves)
    int blocks = (nblk + 7) / 8;             // 8 waves / 256-thread block
    gat_gemm_wmma<<<blocks, 256, 0, stream>>>(h, Ww, Wb, data, N);
  }
  {
    int blocks = (N + 7) / 8;                // one wave per node
    gat_st<<<blocks, 256, 0, stream>>>(data, a, sv, tv, N);
  }
  {
    int blocks = (int)((E + 7) / 8);         // one wave per edge
    gat_edge<<<blocks, 256, 0, stream>>>(eg, eg + E, sv, tv, data, rowsum, out, E);
  }
  {
    long n = (long)N * 16;                   // 4 cols per thread
    int blocks = (int)((n + 255) / 256);
    gat_fin<<<blocks, 256, 0, stream>>>(out, data, rowsum, N);
  }
}